// ROF_GCNConv_11682311045368
// MI455X (gfx1250) — compile-verified
//
#include <hip/hip_runtime.h>

typedef float v2f __attribute__((ext_vector_type(2)));
typedef float v8f __attribute__((ext_vector_type(8)));

#define N_NODES 100000
#define N_EDGES 1600000
#define IN_C 64
#define OUT_C 64

#define ROW_TILES (N_NODES / 16)          // 6250 (exact)
#define WAVES_PER_BLOCK 8
#define EDGES_PER_THREAD 128
#define N_CHUNKS (N_EDGES / EDGES_PER_THREAD)  // 12500 (exact)

// ---------------------------------------------------------------------------
// Kernel 1: h = x @ W   via V_WMMA_F32_16X16X4_F32
// One wave computes a 16x64 tile of h (four 16x16 WMMA accumulators).
// ---------------------------------------------------------------------------
__global__ __launch_bounds__(256) void gemm_xw_kernel(
    const float* __restrict__ x,   // [N_NODES, IN_C]
    const float* __restrict__ w,   // [IN_C, OUT_C]
    float* __restrict__ h)         // [N_NODES, OUT_C]
{
    const int wave = threadIdx.x >> 5;
    const int lane = threadIdx.x & 31;
    const int tile = blockIdx.x * WAVES_PER_BLOCK + wave;
    if (tile >= ROW_TILES) return;      // wave-uniform: EXEC stays all-1s for WMMA

    const int half = lane >> 4;         // 0: lanes 0-15, 1: lanes 16-31
    const int l    = lane & 15;
    const int row  = tile * 16 + l;     // M index held by this lane (A matrix)

    v8f acc0 = {}, acc1 = {}, acc2 = {}, acc3 = {};

    for (int k0 = 0; k0 < IN_C; k0 += 4) {
        // A fragment 16x4: lanes 0-15 hold K=k0..k0+1, lanes 16-31 hold K=k0+2..k0+3
        v2f a;
        a.x = x[row * IN_C + k0 + half * 2 + 0];
        a.y = x[row * IN_C + k0 + half * 2 + 1];

        // B fragment 4x16 per N-tile: VGPR0 = row (k0+half*2), VGPR1 = row (k0+half*2+1)
        const float* wr0 = &w[(k0 + half * 2 + 0) * OUT_C];
        const float* wr1 = &w[(k0 + half * 2 + 1) * OUT_C];
        v2f b0, b1, b2, b3;
        b0.x = wr0[ 0 + l];  b0.y = wr1[ 0 + l];
        b1.x = wr0[16 + l];  b1.y = wr1[16 + l];
        b2.x = wr0[32 + l];  b2.y = wr1[32 + l];
        b3.x = wr0[48 + l];  b3.y = wr1[48 + l];

        acc0 = __builtin_amdgcn_wmma_f32_16x16x4_f32(false, a, false, b0, (short)0, acc0, false, false);
        acc1 = __builtin_amdgcn_wmma_f32_16x16x4_f32(false, a, false, b1, (short)0, acc1, false, false);
        acc2 = __builtin_amdgcn_wmma_f32_16x16x4_f32(false, a, false, b2, (short)0, acc2, false, false);
        acc3 = __builtin_amdgcn_wmma_f32_16x16x4_f32(false, a, false, b3, (short)0, acc3, false, false);
    }

    // D layout (16x16 f32): VGPR r -> M = r + half*8, N = l
    #pragma unroll
    for (int r = 0; r < 8; ++r) {
        float* hp = &h[(size_t)(tile * 16 + half * 8 + r) * OUT_C];
        hp[ 0 + l] = acc0[r];
        hp[16 + l] = acc1[r];
        hp[32 + l] = acc2[r];
        hp[48 + l] = acc3[r];
    }
}

// ---------------------------------------------------------------------------
// Kernel 0: zero the accumulation buffer (deterministic across graph replays)
// ---------------------------------------------------------------------------
__global__ __launch_bounds__(256) void zero_kernel(float* __restrict__ p, int n)
{
    int i = blockIdx.x * blockDim.x + threadIdx.x;
    if (i < n) p[i] = 0.0f;
}

// ---------------------------------------------------------------------------
// Kernel 2: segment-sum gather. row_index is sorted, so each thread owns a
// contiguous run of 128 edges x 4 channels, accumulates locally while the
// destination row repeats, and flushes with f32 atomics on row change.
// Lanes 0-15 of a wave share an edge chunk across all 64 channels, so the
// h[col] gathers coalesce to full 256B lines (L2 resident: h is 25.6MB).
// ---------------------------------------------------------------------------
__global__ __launch_bounds__(256) void aggregate_kernel(
    const float* __restrict__ h,          // [N_NODES, OUT_C]
    const int* __restrict__ column_index, // [N_EDGES]
    const int* __restrict__ row_index,    // [N_EDGES] sorted
    float* __restrict__ tmp)              // [N_NODES, OUT_C] pre-zeroed
{
    const int tid   = blockIdx.x * blockDim.x + threadIdx.x;
    const int chunk = tid >> 4;           // 16 channel-groups per chunk
    const int cg    = tid & 15;           // channels [cg*4, cg*4+4)
    if (chunk >= N_CHUNKS) return;

    const int e0 = chunk * EDGES_PER_THREAD;
    int cur = row_index[e0];
    float a0 = 0.f, a1 = 0.f, a2 = 0.f, a3 = 0.f;

    for (int i = 0; i < EDGES_PER_THREAD; ++i) {
        const int e = e0 + i;
        const int r = row_index[e];
        if (r != cur) {
            float* dst = &tmp[(size_t)cur * OUT_C + cg * 4];
            atomicAdd(dst + 0, a0);
            atomicAdd(dst + 1, a1);
            atomicAdd(dst + 2, a2);
            atomicAdd(dst + 3, a3);
            cur = r;
            a0 = a1 = a2 = a3 = 0.f;
        }
        const int c = column_index[e];
        const float4 hv = *(const float4*)(h + (size_t)c * OUT_C + cg * 4);
        a0 += hv.x; a1 += hv.y; a2 += hv.z; a3 += hv.w;
    }
    float* dst = &tmp[(size_t)cur * OUT_C + cg * 4];
    atomicAdd(dst + 0, a0);
    atomicAdd(dst + 1, a1);
    atomicAdd(dst + 2, a2);
    atomicAdd(dst + 3, a3);
}

// ---------------------------------------------------------------------------
// Kernel 3: out = tmp * degrees[:,None] + bias
// ---------------------------------------------------------------------------
__global__ __launch_bounds__(256) void finalize_kernel(
    const float* __restrict__ tmp,
    const float* __restrict__ degrees,
    const float* __restrict__ bias,
    float* __restrict__ out)
{
    int i = blockIdx.x * blockDim.x + threadIdx.x;
    if (i < N_NODES * OUT_C) {
        const int n = i / OUT_C;
        const int c = i & (OUT_C - 1);
        out[i] = tmp[i] * degrees[n] + bias[c];
    }
}

extern "C" void kernel_launch(void* const* d_in, const int* in_sizes, int n_in,
                              void* d_out, int out_size, void* d_ws, size_t ws_size,
                              hipStream_t stream)
{
    const float* x            = (const float*)d_in[0];
    const float* weight       = (const float*)d_in[1];
    const float* bias         = (const float*)d_in[2];
    const int*   column_index = (const int*)d_in[3];
    const int*   row_index    = (const int*)d_in[4];
    const float* degrees      = (const float*)d_in[5];
    float*       out          = (float*)d_out;

    float* h   = (float*)d_ws;                       // 25.6 MB
    float* tmp = h + (size_t)N_NODES * OUT_C;        // 25.6 MB

    const int NC = N_NODES * OUT_C;                  // 6.4M elements

    zero_kernel<<<(NC + 255) / 256, 256, 0, stream>>>(tmp, NC);

    gemm_xw_kernel<<<(ROW_TILES + WAVES_PER_BLOCK - 1) / WAVES_PER_BLOCK, 256, 0, stream>>>(
        x, weight, h);

    const int agg_threads = N_CHUNKS * 16;           // 200000
    aggregate_kernel<<<(agg_threads + 255) / 256, 256, 0, stream>>>(
        h, column_index, row_index, tmp);

    finalize_kernel<<<(NC + 255) / 256, 256, 0, stream>>>(tmp, degrees, bias, out);
}